// LangModel_34943853920888
// MI455X (gfx1250) — compile-verified
//
#include <hip/hip_runtime.h>
#include <math.h>

// ---------------------------------------------------------------------------
// Batched tiny-RNN (HIDDEN=30, 5 steps, 12 classes) on gfx1250 via fp32 WMMA.
//
// D = A x B + C with V_WMMA_F32_16X16X4_F32:
//   A: 16x4 f32, lane l<16 holds A[l][k0], A[l][k0+1] (k0 = 4t), lane l>=16
//      holds A[l-16][k0+2], A[l-16][k0+3]  -> v2f per lane.
//   B: 4x16 f32, lane l holds B[k][n] with n = l%16, k = 4t + 2*(l/16)+{0,1}.
//   C/D: 16x16 f32, lane l holds rows (l/16)*8 + r (r=0..7), col l%16 -> v8f.
//
// Elementwise math uses gfx1250 HW transcendentals (v_tanh_f32, v_exp_f32,
// v_log_f32) instead of branchy libm expansions; TRANS ops co-execute with
// the WMMA pipe. All K/N padding handled by zero-padded LDS weight images.
// ---------------------------------------------------------------------------

typedef __attribute__((ext_vector_type(2))) float v2f;
typedef __attribute__((ext_vector_type(4))) float v4f;
typedef __attribute__((ext_vector_type(8))) float v8f;

#define WAVES_PER_BLOCK 8
#define TILE_M 16
#define KT 8                 // ceil(32/4) K-tiles of the padded K=32 dimension
#define MAX_ITER 5
#define NCLS 12

#define LOG2E 1.4426950408889634f
#define LN2   0.6931471805599453f

#if __has_builtin(__builtin_amdgcn_tanhf)
__device__ __forceinline__ float fast_tanh(float x) {
  return __builtin_amdgcn_tanhf(x);      // v_tanh_f32 (gfx1250 HW trans op)
}
#else
__device__ __forceinline__ float fast_tanh(float x) { return tanhf(x); }
#endif

__device__ __forceinline__ float fast_exp(float x) {   // e^x, x <= 0 here
  return __builtin_amdgcn_exp2f(x * LOG2E);            // v_exp_f32 = 2^x
}
__device__ __forceinline__ float fast_log(float x) {   // ln(x), x >= 1 here
  return __builtin_amdgcn_logf(x) * LN2;               // v_log_f32 = log2(x)
}

__device__ __forceinline__ v8f wmma_f32_k4(v2f a, v2f b, v8f c) {
  // 8 args: (neg_a, A, neg_b, B, c_mod, C, reuse_a, reuse_b)
  return __builtin_amdgcn_wmma_f32_16x16x4_f32(false, a, false, b, (short)0, c,
                                               false, false);
}

__global__ __launch_bounds__(256) void rnn_wmma_kernel(
    const float* __restrict__ x,
    const float* __restrict__ W_ih, const float* __restrict__ b_ih,
    const float* __restrict__ W_hh, const float* __restrict__ b_hh,
    const float* __restrict__ W_fc, const float* __restrict__ b_fc,
    float* __restrict__ y, float* __restrict__ hout, int N) {
  // ---- LDS -----------------------------------------------------------------
  __shared__ float sWhh[32][32];                 // [n][k], zero padded (4 KB)
  __shared__ float sWfc[16][32];                 // [n][k], zero padded (2 KB)
  __shared__ float sWih[32];                     // padded
  __shared__ float sB[32];                       // b_ih + b_hh, padded
  __shared__ float sBfc[16];                     // padded
  __shared__ float sH[WAVES_PER_BLOCK][16][32];  // per-wave h tile (16 KB)
  __shared__ float sL[WAVES_PER_BLOCK][16][16];  // per-wave logits (8 KB)

  const int tid  = threadIdx.x;
  const int lane = tid & 31;
  const int wave = tid >> 5;

  // ---- stage weights (cooperative, once per block) -------------------------
  for (int i = tid; i < 32 * 32; i += 256) {
    int n = i >> 5, k = i & 31;
    sWhh[n][k] = (n < 30 && k < 30) ? W_hh[n * 30 + k] : 0.0f;
  }
  for (int i = tid; i < 16 * 32; i += 256) {
    int n = i >> 5, k = i & 31;
    sWfc[n][k] = (n < 12 && k < 30) ? W_fc[n * 30 + k] : 0.0f;
  }
  if (tid < 32) {
    sWih[tid] = (tid < 30) ? W_ih[tid] : 0.0f;
    sB[tid]   = (tid < 30) ? (b_ih[tid] + b_hh[tid]) : 0.0f;
  }
  if (tid < 16) sBfc[tid] = (tid < 12) ? b_fc[tid] : 0.0f;
  __syncthreads();

  // ---- per-lane WMMA fragment coordinates ----------------------------------
  const int col   = lane & 15;          // B/D column (N index within tile)
  const int halfK = (lane >> 4) << 1;   // 0 for lanes 0-15, 2 for lanes 16-31
  const int rbase = (lane >> 4) << 3;   // D rows: 0..7 or 8..15

  // ---- resident B fragments (weights) --------------------------------------
  v2f bhh0[KT], bhh1[KT], bfcf[KT];
#pragma unroll
  for (int t = 0; t < KT; ++t) {
    const int k0 = 4 * t + halfK;       // even -> 8B aligned
    bhh0[t] = *(const v2f*)&sWhh[col][k0];        // N-tile 0 (cols 0..15)
    bhh1[t] = *(const v2f*)&sWhh[col + 16][k0];   // N-tile 1 (cols 16..31)
    bfcf[t] = *(const v2f*)&sWfc[col][k0];        // FC (cols 0..15)
  }
  const float biasH0 = sB[col];
  const float biasH1 = sB[col + 16];
  const float biasFC = sBfc[col];

  // ---- this wave's 16-sample tile ------------------------------------------
  const int base = (blockIdx.x * WAVES_PER_BLOCK + wave) * TILE_M;

  // step 0: h = tanh(x * W_ih + b_ih + b_hh)   (h0 == 0, no matmul needed)
  {
    const float wv = sWih[lane];
    const float bv = sB[lane];
#pragma unroll
    for (int r = 0; r < TILE_M; ++r) {
      const float xv = (base + r < N) ? x[base + r] : 0.0f;
      sH[wave][r][lane] = fast_tanh(fmaf(xv, wv, bv));  // cols>=30 -> 0
    }
  }

  v2f afrag[KT];
  const int arow = lane & 15;

#pragma unroll 1
  for (int iter = 0; iter < MAX_ITER; ++iter) {
    // ---- recurrent update (iters 1..4): h = tanh(h @ W_hh^T + b) ----------
    if (iter > 0) {
#pragma unroll
      for (int t = 0; t < KT; ++t)
        afrag[t] = *(const v2f*)&sH[wave][arow][4 * t + halfK];

      v8f acc0 = {}, acc1 = {};
#pragma unroll
      for (int t = 0; t < KT; ++t) {
        acc0 = wmma_f32_k4(afrag[t], bhh0[t], acc0);
        acc1 = wmma_f32_k4(afrag[t], bhh1[t], acc1);
      }
#pragma unroll
      for (int r = 0; r < 8; ++r) {
        sH[wave][rbase + r][col]      = fast_tanh(acc0[r] + biasH0);
        sH[wave][rbase + r][col + 16] = fast_tanh(acc1[r] + biasH1);
      }
    }

    // ---- FC: logits = h @ W_fc^T + b_fc -----------------------------------
#pragma unroll
    for (int t = 0; t < KT; ++t)
      afrag[t] = *(const v2f*)&sH[wave][arow][4 * t + halfK];

    v8f accL = {};
#pragma unroll
    for (int t = 0; t < KT; ++t) accL = wmma_f32_k4(afrag[t], bfcf[t], accL);
#pragma unroll
    for (int r = 0; r < 8; ++r)
      sL[wave][rbase + r][col] = accL[r] + biasFC;

    // ---- log_softmax over 12 classes, one row per lane (lanes 0..15) ------
    if (lane < TILE_M && base + lane < N) {
      const float* lg = &sL[wave][lane][0];
      float v[NCLS];
#pragma unroll
      for (int c = 0; c < NCLS; ++c) v[c] = lg[c];
      float m = v[0];
#pragma unroll
      for (int c = 1; c < NCLS; ++c) m = fmaxf(m, v[c]);
      float s = 0.0f;
#pragma unroll
      for (int c = 0; c < NCLS; ++c) s += fast_exp(v[c] - m);
      const float lse = m + fast_log(s);

      // 12 outputs per row, 16B aligned (row stride 240B, iter stride 48B):
      // emit as 3 x global_store_b128.
      float* yo = y + (size_t)(base + lane) * (MAX_ITER * NCLS) + iter * NCLS;
      v4f p0 = {v[0] - lse, v[1] - lse, v[2] - lse, v[3] - lse};
      v4f p1 = {v[4] - lse, v[5] - lse, v[6] - lse, v[7] - lse};
      v4f p2 = {v[8] - lse, v[9] - lse, v[10] - lse, v[11] - lse};
      *(v4f*)(yo + 0) = p0;
      *(v4f*)(yo + 4) = p1;
      *(v4f*)(yo + 8) = p2;
    }
    // reconverged: EXEC is all-1s again before the next iteration's WMMAs
  }

  // ---- final hidden state, fully coalesced (480 contiguous floats/tile) ----
  for (int i = lane; i < TILE_M * 30; i += 32) {
    const int r = i / 30, c = i % 30;
    if (base + r < N) hout[(size_t)base * 30 + i] = sH[wave][r][c];
  }
}

extern "C" void kernel_launch(void* const* d_in, const int* in_sizes, int n_in,
                              void* d_out, int out_size, void* d_ws,
                              size_t ws_size, hipStream_t stream) {
  const float* x    = (const float*)d_in[0];
  const float* W_ih = (const float*)d_in[1];
  const float* b_ih = (const float*)d_in[2];
  const float* W_hh = (const float*)d_in[3];
  const float* b_hh = (const float*)d_in[4];
  const float* W_fc = (const float*)d_in[5];
  const float* b_fc = (const float*)d_in[6];

  const int N = in_sizes[0];           // x is [N,1]
  float* y    = (float*)d_out;                          // [N, 5, 12]
  float* hout = y + (size_t)N * MAX_ITER * NCLS;        // [N, 30]

  const int samples_per_block = WAVES_PER_BLOCK * TILE_M;  // 128
  const int blocks = (N + samples_per_block - 1) / samples_per_block;

  rnn_wmma_kernel<<<blocks, 256, 0, stream>>>(x, W_ih, b_ih, W_hh, b_hh, W_fc,
                                              b_fc, y, hout, N);
}